// CRF_55817394979316
// MI455X (gfx1250) — compile-verified
//
#include <hip/hip_runtime.h>
#include <hip/hip_bf16.h>

#define CRF_B 4096
#define CRF_L 512
#define CRF_T 10
#define CRF_PAD 9

typedef __attribute__((ext_vector_type(2))) float v2f;
typedef __attribute__((ext_vector_type(8))) float v8f;

// ---------------------------------------------------------------------------
// Kernel 1: fused Viterbi forward + backtrack + gold-path numerator.
// One thread per sequence (4096 threads). trans/start/end are indexed with
// compile-time constants only (dynamic gathers go straight to global, cached),
// so they live in scalar/vector registers, not scratch. Backpointers are
// packed 10x4 bits into one u64 per (b,t) in d_ws.
// ---------------------------------------------------------------------------
__global__ __launch_bounds__(256) void crf_viterbi_num_kernel(
    const float* __restrict__ em, const int* __restrict__ tags,
    const float* __restrict__ start, const float* __restrict__ endt,
    const float* __restrict__ trans,
    float* __restrict__ out_path, float* __restrict__ num_out,
    unsigned long long* __restrict__ bp, float* __restrict__ loss)
{
    if (blockIdx.x == 0 && threadIdx.x == 0) *loss = 0.0f;  // K2 runs after us

    const int b = blockIdx.x * blockDim.x + threadIdx.x;

    // Constant-indexed copies -> stay in registers (uniform addresses).
    float tr[CRF_T][CRF_T];
#pragma unroll
    for (int i = 0; i < CRF_T; ++i)
#pragma unroll
        for (int j = 0; j < CRF_T; ++j) tr[i][j] = trans[i * CRF_T + j];
    float st[CRF_T], en[CRF_T];
#pragma unroll
    for (int j = 0; j < CRF_T; ++j) { st[j] = start[j]; en[j] = endt[j]; }

    const float* emb = em + (size_t)b * (CRF_L * CRF_T);
    const int* tb = tags + (size_t)b * CRF_L;
    unsigned long long* bpb = bp + (size_t)b * CRF_L;

    const int tag0 = tb[0];                       // len >= 1, always valid
    float sc[CRF_T];
#pragma unroll
    for (int j = 0; j < CRF_T; ++j) sc[j] = st[j] + emb[j];
    float nscore = start[tag0] + emb[tag0];       // dynamic idx -> global (cached)
    int prev = tag0;

    for (int t = 1; t < CRF_L; ++t) {
        const float* emt = emb + t * CRF_T;
        __builtin_prefetch(emt + 4 * CRF_T, 0, 1);   // global_prefetch_b8
        const int cur = tb[t];
        if (cur != CRF_PAD) {
            // numerator step
            nscore += trans[prev * CRF_T + cur] + emt[cur];
            prev = cur;
            // viterbi step (first-max argmax, matching jnp.argmax)
            unsigned long long pack = 0ull;
            float ns[CRF_T];
#pragma unroll
            for (int j = 0; j < CRF_T; ++j) {
                float best = sc[0] + tr[0][j];
                int arg = 0;
#pragma unroll
                for (int i = 1; i < CRF_T; ++i) {
                    float c = sc[i] + tr[i][j];
                    if (c > best) { best = c; arg = i; }
                }
                ns[j] = best + emt[j];
                pack |= ((unsigned long long)arg) << (4 * j);
            }
#pragma unroll
            for (int j = 0; j < CRF_T; ++j) sc[j] = ns[j];
            bpb[t] = pack;
        } else {
            bpb[t] = 0x9876543210ull;   // identity backpointers (masked step)
        }
    }
    num_out[b] = nscore + endt[prev];   // prev == tag at last valid position

    // last tag = argmax(final + end)
    float bestf = sc[0] + en[0];
    int last = 0;
#pragma unroll
    for (int j = 1; j < CRF_T; ++j) {
        float c = sc[j] + en[j];
        if (c > bestf) { bestf = c; last = j; }
    }

    // backtrack (same thread: our own prior global stores, program order is fine)
    float* op = out_path + (size_t)b * CRF_L;
    int curt = last;
    for (int t = CRF_L - 1; t >= 1; --t) {
        op[t] = (tb[t] != CRF_PAD) ? (float)curt : 0.0f;
        curt = (int)((bpb[t] >> (4 * curt)) & 0xFull);
    }
    op[0] = (float)curt;                // position 0 always unmasked (len >= 1)
}

// ---------------------------------------------------------------------------
// Kernel 2: log-partition via WMMA in the exp domain.
// One wave handles a tile of 16 sequences. State kept as scaled probabilities
// P^T (M=state j, N=batch) in the 16x16 f32 C layout (8 VGPRs):
//   C reg r, lane l: state = r + 8*(l>=16), batch = l&15.
// Step: P' = Ehat x P (Ehat = exp(trans), constant A operand), then
// elementwise * exp(em_t), per-batch mask blend, per-batch rescale by the max
// with running log-scale M. K=16 in 4 chunks of V_WMMA_F32_16X16X4_F32;
// B chunk c is literally (C[2c], C[2c+1]): its K-slot(v,half)=2*half+v maps to
// state 2c+v+8*half, and A is built with the same slot->state map.
// Padded states (>=10) have zero A rows/cols, so their lanes stay exactly 0 --
// which also means the emission value fed to those lanes is irrelevant, so the
// loads use a clamped (always in-bounds) index instead of a predicated load.
// ---------------------------------------------------------------------------
__global__ __launch_bounds__(256) void crf_partition_kernel(
    const float* __restrict__ em, const int* __restrict__ tags,
    const float* __restrict__ start, const float* __restrict__ endt,
    const float* __restrict__ trans,
    const float* __restrict__ numb, float* __restrict__ loss)
{
    const int lane = threadIdx.x & 31;
    const int wave = threadIdx.x >> 5;
    const int tile = blockIdx.x * 8 + wave;     // 0..255
    const int hi = lane >> 4;                   // which lane half
    const int b = tile * 16 + (lane & 15);      // this lane's batch (B/C/D N idx)
    const int jA = lane & 15;                   // this lane's output state (A M idx)

    const float LOG2E = 1.4426950408889634f;
    const float LN2   = 0.6931471805599453f;

    // Constant A operand: A[c] holds Ehat[i][jA] with i = 2c+v+8*hi.
    v2f A[4];
#pragma unroll
    for (int c = 0; c < 4; ++c) {
#pragma unroll
        for (int v = 0; v < 2; ++v) {
            const int i = 2 * c + v + 8 * hi;
            float val = 0.0f;
            if (i < CRF_T && jA < CRF_T) val = expf(trans[i * CRF_T + jA]);
            A[c][v] = val;
        }
    }

    // init: p_j = exp(start_j + em[b,0,j] - M),  M = row max
    const float* row0 = em + (size_t)b * (CRF_L * CRF_T);
    float lg[8];
#pragma unroll
    for (int r = 0; r < 8; ++r) {
        const int jj = r + 8 * hi;
        lg[r] = (jj < CRF_T) ? (start[jj] + row0[jj]) : -1e30f;
    }
    float M = lg[0];
#pragma unroll
    for (int r = 1; r < 8; ++r) M = fmaxf(M, lg[r]);
    M = fmaxf(M, __shfl_xor(M, 16));            // combine the two state halves
    v8f p;
#pragma unroll
    for (int r = 0; r < 8; ++r) p[r] = expf(lg[r] - M);

    const int* tcol = tags + (size_t)b * CRF_L;

    for (int t = 1; t < CRF_L; ++t) {
        const float* rowp = em + ((size_t)b * CRF_L + t) * CRF_T;
        __builtin_prefetch(rowp + CRF_T, 0, 1);

        // branch-free emission loads (clamped index, always in-bounds)
        float ev[8];
#pragma unroll
        for (int r = 0; r < 8; ++r) {
            int jj = r + 8 * hi;
            jj = (jj < CRF_T) ? jj : (CRF_T - 1);
            ev[r] = rowp[jj];
        }
        const int cur = tcol[t];
        const bool m = (cur != CRF_PAD);        // uniform per batch -> per lane

        v8f q = {0.f, 0.f, 0.f, 0.f, 0.f, 0.f, 0.f, 0.f};
#pragma unroll
        for (int c = 0; c < 4; ++c) {
            v2f Bc;
            Bc[0] = p[2 * c];
            Bc[1] = p[2 * c + 1];
            q = __builtin_amdgcn_wmma_f32_16x16x4_f32(
                    false, A[c], false, Bc, (short)0, q, false, false);
        }

        float s = 0.0f;
        v8f pn;
#pragma unroll
        for (int r = 0; r < 8; ++r) {
            // exp(em) via hardware v_exp_f32; |em| is O(5), well inside range
            const float cand = q[r] * __builtin_amdgcn_exp2f(ev[r] * LOG2E);
            pn[r] = m ? cand : p[r];
            s = fmaxf(s, pn[r]);
        }
        s = fmaxf(s, __shfl_xor(s, 16));        // per-batch max over 16 states
        const float rs = __builtin_amdgcn_rcpf(s);       // v_rcp_f32
        M -= __builtin_amdgcn_logf(rs) * LN2;            // exactly -log(rs)
#pragma unroll
        for (int r = 0; r < 8; ++r) p[r] = pn[r] * rs;
    }

    // den = M + log(sum_j p_j * exp(end_j))
    float acc = 0.0f;
#pragma unroll
    for (int r = 0; r < 8; ++r) {
        const int jj = r + 8 * hi;
        const float ee = (jj < CRF_T) ? expf(endt[jj]) : 0.0f;
        acc += p[r] * ee;
    }
    acc += __shfl_xor(acc, 16);
    const float den = M + logf(acc);

    // loss = sum_b (den - num); each batch appears in two lanes -> halve.
    float diff = den - numb[b];
#pragma unroll
    for (int o = 1; o < 32; o <<= 1) diff += __shfl_xor(diff, o);
    if (lane == 0) atomicAdd(loss, diff * 0.5f);
}

// ---------------------------------------------------------------------------
extern "C" void kernel_launch(void* const* d_in, const int* in_sizes, int n_in,
                              void* d_out, int out_size, void* d_ws, size_t ws_size,
                              hipStream_t stream) {
    const float* fc    = (const float*)d_in[0];   // (B, L, T) f32
    const int*   tags  = (const int*)d_in[1];     // (B, L) i32
    const float* start = (const float*)d_in[2];   // (T,)
    const float* endt  = (const float*)d_in[3];   // (T,)
    const float* trans = (const float*)d_in[4];   // (T, T)

    float* out  = (float*)d_out;                  // [B*L path floats][1 loss]
    float* loss = out + (size_t)CRF_B * CRF_L;

    unsigned long long* bp = (unsigned long long*)d_ws;            // B*L u64
    float* numb = (float*)((char*)d_ws +
                           (size_t)CRF_B * CRF_L * sizeof(unsigned long long));

    // K1: viterbi + numerator (also zeroes the loss accumulator)
    crf_viterbi_num_kernel<<<CRF_B / 256, 256, 0, stream>>>(
        fc, tags, start, endt, trans, out, numb, bp, loss);

    // K2: WMMA exp-domain partition + loss accumulation (stream-ordered after K1)
    crf_partition_kernel<<<(CRF_B / 16) / 8, 256, 0, stream>>>(
        fc, tags, start, endt, trans, numb, loss);
}